// Model_64355789964011
// MI455X (gfx1250) — compile-verified
//
#include <hip/hip_runtime.h>
#include <math.h>

typedef __attribute__((ext_vector_type(2))) float v2f;
typedef __attribute__((ext_vector_type(8))) float v8f;

#define NB   4
#define NN   512
#define LL   512
#define HH   8
#define DD   64
#define TAUD 4
#define EE   25
#define KK   32
#define LOUT 416

// ---------------------------------------------------------------------------
// K1: fused dilated-conv1d + Linear(L_OUT->1) + exact GELU.
//   y[h,e]   = sum_t proj_w[t] * x[bn, t+4e, h]        (proj commuted into conv)
//   embed[d] = gelu(sum_{h,e} conv_w[d,h,e]*y[h,e] + b)
// One workgroup per bn; x row staged once through LDS (single pass over 33.5MB).
// ---------------------------------------------------------------------------
__global__ void embed_kernel(const float* __restrict__ x,
                             const float* __restrict__ conv_w,
                             const float* __restrict__ proj_w,
                             const float* __restrict__ proj_b,
                             float* __restrict__ embed) {
  __shared__ float xs[HH * LL];     // 16 KB, xs[h*LL + l]
  __shared__ float pw[LOUT];
  __shared__ float yb[HH * EE];     // 200
  const int bn  = blockIdx.x;
  const int tid = threadIdx.x;
  const float* xp = x + (size_t)bn * LL * HH;   // x[bn, l, h] contiguous in h
  for (int i = tid; i < LL * HH; i += 256) {
    int l = i >> 3, h = i & 7;
    xs[h * LL + l] = xp[i];
  }
  for (int i = tid; i < LOUT; i += 256) pw[i] = proj_w[i];
  __syncthreads();
  if (tid < HH * EE) {
    int h = tid / EE, e = tid % EE;
    const float* xr = &xs[h * LL + e * TAUD];
    float acc = 0.f;
    #pragma unroll 4
    for (int t = 0; t < LOUT; ++t) acc = fmaf(pw[t], xr[t], acc);
    yb[tid] = acc;
  }
  __syncthreads();
  if (tid < DD) {
    const float* w = conv_w + (size_t)tid * (HH * EE);  // conv_w[d,h,e]
    float acc = proj_b[0];
    #pragma unroll 4
    for (int j = 0; j < HH * EE; ++j) acc = fmaf(w[j], yb[j], acc);
    float g = 0.5f * acc * (1.0f + erff(acc * 0.70710678118654752440f));
    embed[(size_t)bn * DD + tid] = g;
  }
}

// ---------------------------------------------------------------------------
// K2a: v = target - rowmean(target).  One block per row.
// ---------------------------------------------------------------------------
__global__ void center_kernel(const float* __restrict__ target,
                              float* __restrict__ v) {
  __shared__ float red[256];
  const int row = blockIdx.x, tid = threadIdx.x;
  const float* t = target + (size_t)row * LL;
  float a = t[tid], b = t[tid + 256];
  red[tid] = a + b;
  __syncthreads();
  for (int s = 128; s > 0; s >>= 1) {
    if (tid < s) red[tid] += red[tid + s];
    __syncthreads();
  }
  const float mean = red[0] * (1.0f / (float)LL);
  v[(size_t)row * LL + tid]       = a - mean;
  v[(size_t)row * LL + tid + 256] = b - mean;
}

// ---------------------------------------------------------------------------
// WMMA fp32 16x16x4 fragment helpers (wave32, one wave per block).
//   A 16x4 f32: lanes 0-15 hold K{0,1}, lanes 16-31 hold K{2,3}, M = lane&15
//   B  4x16 f32: mirrored with N on lanes
//   C/D: VGPR i -> (M = i + 8*(lane>=16), N = lane&15)
// ---------------------------------------------------------------------------

// K2b: vx = v @ W0, vh = v @ U0.  grid = (BN/16, KK/16, 2), block = 32.
__global__ void gemm_vw_kernel(const float* __restrict__ v,
                               const float* __restrict__ W0,
                               const float* __restrict__ U0,
                               float* __restrict__ vx,
                               float* __restrict__ vh) {
  const int lane = threadIdx.x;
  const int half = lane >> 4, mn = lane & 15, kd = half * 2;
  const int r0 = blockIdx.x * 16;          // row tile over B*N
  const int c0 = blockIdx.y * 16;          // col tile over K=32
  const float* Bm = blockIdx.z ? U0 : W0;  // (L, K) row-major
  float* out      = blockIdx.z ? vh : vx;
  const float* arow = v + (size_t)(r0 + mn) * LL;
  v8f acc = {};
  for (int k = 0; k < LL; k += 4) {
    v2f a = *(const v2f*)(arow + k + kd);
    v2f b;
    b[0] = Bm[(size_t)(k + kd)     * KK + c0 + mn];
    b[1] = Bm[(size_t)(k + kd + 1) * KK + c0 + mn];
    acc = __builtin_amdgcn_wmma_f32_16x16x4_f32(false, a, false, b,
                                                (short)0, acc, false, false);
  }
  #pragma unroll
  for (int i = 0; i < 8; ++i) {
    int row = r0 + half * 8 + i;
    out[(size_t)row * KK + c0 + mn] = acc[i];
  }
}

// K2c: inverse norms 1/sqrt(sumsq + eps).
__global__ void norm_kernel(const float* __restrict__ vx,
                            const float* __restrict__ vh,
                            float* __restrict__ inx,
                            float* __restrict__ inh) {
  const int r = blockIdx.x * 256 + threadIdx.x;   // 0..2047
  float sx = 0.f, sh = 0.f;
  const float* a = vx + (size_t)r * KK;
  const float* b = vh + (size_t)r * KK;
  #pragma unroll
  for (int k = 0; k < KK; ++k) { sx = fmaf(a[k], a[k], sx); sh = fmaf(b[k], b[k], sh); }
  inx[r] = 1.0f / sqrtf(sx + 1e-6f);
  inh[r] = 1.0f / sqrtf(sh + 1e-6f);
}

// K3: pearson[b,n,m] = (vx[b,n,:] . vh[b,m,:]) * inx[b,n] * inh[b,m]
// grid = (N/16, N/16, B), block = 32.
__global__ void pearson_kernel(const float* __restrict__ vx,
                               const float* __restrict__ vh,
                               const float* __restrict__ inx,
                               const float* __restrict__ inh,
                               float* __restrict__ pearson) {
  const int lane = threadIdx.x;
  const int half = lane >> 4, mn = lane & 15, kd = half * 2;
  const int b  = blockIdx.z;
  const int n0 = blockIdx.x * 16, m0 = blockIdx.y * 16;
  const float* ax = vx + ((size_t)b * NN + n0 + mn) * KK;
  const float* bx = vh + ((size_t)b * NN + m0 + mn) * KK;  // B[k][n]=vh[m0+n][k]
  v8f acc = {};
  #pragma unroll
  for (int k = 0; k < KK; k += 4) {
    v2f a  = *(const v2f*)(ax + k + kd);
    v2f bb = *(const v2f*)(bx + k + kd);
    acc = __builtin_amdgcn_wmma_f32_16x16x4_f32(false, a, false, bb,
                                                (short)0, acc, false, false);
  }
  const float sh = inh[(size_t)b * NN + m0 + mn];
  #pragma unroll
  for (int i = 0; i < 8; ++i) {
    int row = n0 + half * 8 + i, col = m0 + mn;
    float pv = acc[i] * inx[(size_t)b * NN + row] * sh;
    pearson[((size_t)b * NN + row) * NN + col] = pv;
  }
}

// K5: attn = mean over b of pearson -> d_out + B*N*D
__global__ void attn_kernel(const float* __restrict__ pearson,
                            float* __restrict__ attn) {
  const size_t i  = (size_t)blockIdx.x * 256 + threadIdx.x;  // 0..NN*NN-1
  const size_t P  = (size_t)NN * NN;
  attn[i] = 0.25f * (pearson[i] + pearson[i + P] + pearson[i + 2 * P] + pearson[i + 3 * P]);
}

// K4: per row, top-32 (lowest-index tie-break), scatter, softmax over full row.
// One block (256 threads) per row; 32 iterative argmax passes in LDS.
__global__ void topk_softmax_kernel(const float* __restrict__ pearson,
                                    float* __restrict__ mult) {
  __shared__ float sv[NN];
  __shared__ float wv[NN];
  __shared__ float rv[256];
  __shared__ int   ri[256];
  __shared__ float sM, sZ;
  const int row = blockIdx.x, tid = threadIdx.x;
  const float* p = pearson + (size_t)row * NN;
  float a0 = p[tid], a1 = p[tid + 256];
  sv[tid] = a0; sv[tid + 256] = a1;
  wv[tid] = a0; wv[tid + 256] = a1;
  __syncthreads();
  float M = 0.f, Zt = 0.f;       // live only in thread 0
  for (int it = 0; it < KK; ++it) {
    float v0 = wv[tid], v1 = wv[tid + 256];
    float bv; int bi;
    if (v1 > v0) { bv = v1; bi = tid + 256; } else { bv = v0; bi = tid; }
    rv[tid] = bv; ri[tid] = bi;
    __syncthreads();
    for (int s = 128; s > 0; s >>= 1) {
      if (tid < s) {
        float ov = rv[tid + s]; int oi = ri[tid + s];
        if (ov > rv[tid] || (ov == rv[tid] && oi < ri[tid])) { rv[tid] = ov; ri[tid] = oi; }
      }
      __syncthreads();
    }
    if (tid == 0) {
      wv[ri[0]] = -1e30f;                      // mark chosen
      if (it == 0) M = fmaxf(rv[0], 0.f);      // row max incl. implicit zeros
      Zt += expf(rv[0] - M);
    }
    __syncthreads();
  }
  if (tid == 0) { sM = M; sZ = Zt + (float)(NN - KK) * expf(-M); }
  __syncthreads();
  const float invZ = 1.0f / sZ;
  const float zval = expf(-sM) * invZ;         // softmax of the scattered zeros
  float* o = mult + (size_t)row * NN;
  o[tid]       = (wv[tid]       == -1e30f) ? expf(sv[tid]       - sM) * invZ : zval;
  o[tid + 256] = (wv[tid + 256] == -1e30f) ? expf(sv[tid + 256] - sM) * invZ : zval;
}

// K6: out[b,n,d] = sum_m mult[b,n,m] * embed[b,m,d].  grid=(N/16, D/16, B).
__global__ void out_gemm_kernel(const float* __restrict__ mult,
                                const float* __restrict__ embed,
                                float* __restrict__ out) {
  const int lane = threadIdx.x;
  const int half = lane >> 4, mn = lane & 15, kd = half * 2;
  const int b  = blockIdx.z;
  const int n0 = blockIdx.x * 16, d0 = blockIdx.y * 16;
  const float* arow = mult + ((size_t)b * NN + n0 + mn) * NN;
  v8f acc = {};
  for (int k = 0; k < NN; k += 4) {
    v2f a = *(const v2f*)(arow + k + kd);
    v2f bb;
    bb[0] = embed[((size_t)b * NN + k + kd)     * DD + d0 + mn];
    bb[1] = embed[((size_t)b * NN + k + kd + 1) * DD + d0 + mn];
    acc = __builtin_amdgcn_wmma_f32_16x16x4_f32(false, a, false, bb,
                                                (short)0, acc, false, false);
  }
  #pragma unroll
  for (int i = 0; i < 8; ++i) {
    int row = n0 + half * 8 + i;
    out[((size_t)b * NN + row) * DD + d0 + mn] = acc[i];
  }
}

// ---------------------------------------------------------------------------
extern "C" void kernel_launch(void* const* d_in, const int* in_sizes, int n_in,
                              void* d_out, int out_size, void* d_ws, size_t ws_size,
                              hipStream_t stream) {
  const float* x      = (const float*)d_in[0];  // (B,N,L,H)
  const float* target = (const float*)d_in[1];  // (B,N,L)
  const float* conv_w = (const float*)d_in[2];  // (D,H,E)
  const float* proj_w = (const float*)d_in[3];  // (L_OUT,)
  const float* proj_b = (const float*)d_in[4];  // scalar
  const float* W0     = (const float*)d_in[5];  // (L,K)
  const float* U0     = (const float*)d_in[6];  // (L,K)

  float* out_p  = (float*)d_out;                      // (B,N,D) = 131072
  float* attn_p = out_p + (size_t)NB * NN * DD;       // (N,N)   = 262144

  // Workspace layout (floats). mult reuses v's buffer (v dead after gemm_vw).
  float* ws      = (float*)d_ws;
  float* v_buf   = ws;                                         // 1,048,576
  float* mult    = ws;                                         // reuse
  float* embed   = ws + 1048576;                               //   131,072
  float* vx      = embed + (size_t)NB * NN * DD;               //    65,536
  float* vh      = vx + (size_t)NB * NN * KK;                  //    65,536
  float* inx     = vh + (size_t)NB * NN * KK;                  //     2,048
  float* inh     = inx + (size_t)NB * NN;                      //     2,048
  float* pearson = inh + (size_t)NB * NN;                      // 1,048,576

  embed_kernel<<<NB * NN, 256, 0, stream>>>(x, conv_w, proj_w, proj_b, embed);
  center_kernel<<<NB * NN, 256, 0, stream>>>(target, v_buf);
  gemm_vw_kernel<<<dim3(NB * NN / 16, KK / 16, 2), 32, 0, stream>>>(v_buf, W0, U0, vx, vh);
  norm_kernel<<<NB * NN / 256, 256, 0, stream>>>(vx, vh, inx, inh);
  pearson_kernel<<<dim3(NN / 16, NN / 16, NB), 32, 0, stream>>>(vx, vh, inx, inh, pearson);
  attn_kernel<<<(NN * NN) / 256, 256, 0, stream>>>(pearson, attn_p);
  topk_softmax_kernel<<<NB * NN, 256, 0, stream>>>(pearson, mult);
  out_gemm_kernel<<<dim3(NN / 16, DD / 16, NB), 32, 0, stream>>>(mult, embed, out_p);
}